// GraphConvSparse_31671088840935
// MI455X (gfx1250) — compile-verified
//
#include <hip/hip_runtime.h>
#include <hip/hip_bf16.h>

// ---------------------------------------------------------------------------
// GraphConv: out = relu(segment_sum(val * (x@W)[col] -> row))
//   x: [100000, 512] f32, W: [512, 128] f32, edges: 3.2M (row, col, val)
//
// MI455X plan:
//   K1: xp = x @ W with V_WMMA_F32_16X16X4_F32 (exact f32 precision),
//       LDS-staged x tile + W k-blocks (bank-conflict padded).
//   SpMM: build CSR on device (hist -> scan -> scatter), then one wave per
//       row gathers xp[col] (L2-resident: xp = 51.2 MB << 192 MB L2),
//       accumulates in VGPRs, fused ReLU, single coalesced store.
//       -> eliminates the 409.6M global f32 atomics of the edge-parallel form.
//   Fallback (if ws too small for CSR): edge-parallel f32-atomic version.
// ---------------------------------------------------------------------------

typedef __attribute__((ext_vector_type(2))) float v2f;
typedef __attribute__((ext_vector_type(8))) float v8f;

#define GC_IN_DIM  512
#define GC_OUT_DIM 128
#define GC_XPAD    (GC_IN_DIM + 4)    // 516: 16 A-rows map to 16 distinct LDS banks
#define GC_WPAD    (GC_OUT_DIM + 16)  // 144: wave halves hit disjoint bank groups

// ---------------------------------------------------------------------------
// K1: dense projection with f32 WMMA.
// Block = 256 threads = 8 waves; rows [r0, r0+16) x all 128 cols.
// ---------------------------------------------------------------------------
__global__ __launch_bounds__(256) void gc_gemm_wmma(const float* __restrict__ x,
                                                    const float* __restrict__ W,
                                                    float* __restrict__ xp,
                                                    int nRows) {
    __shared__ float lx[16 * GC_XPAD];   // 33 KB
    __shared__ float lw[64 * GC_WPAD];   // 36 KB

    const int tid  = threadIdx.x;
    const int r0   = blockIdx.x * 16;
    const int lane = tid & 31;
    const int wave = tid >> 5;
    const int half = lane >> 4;
    const int l16  = lane & 15;
    const int n0   = wave * 16;

    // Stage x tile: 16 rows x 512 = 2048 float4, 8 per thread.
    for (int i = 0; i < 8; ++i) {
        int f4    = tid + i * 256;
        int row   = f4 >> 7;
        int colf4 = f4 & 127;
        float4 v = make_float4(0.f, 0.f, 0.f, 0.f);
        if (r0 + row < nRows)
            v = reinterpret_cast<const float4*>(x + (size_t)(r0 + row) * GC_IN_DIM)[colf4];
        float* dst = &lx[row * GC_XPAD + colf4 * 4];
        dst[0] = v.x; dst[1] = v.y; dst[2] = v.z; dst[3] = v.w;
    }

    v8f acc = {};

    for (int kb = 0; kb < GC_IN_DIM; kb += 64) {
        __syncthreads();
        // Stage W k-block: 64 x 128 = 2048 float4, 8 per thread.
        for (int i = 0; i < 8; ++i) {
            int f4    = tid + i * 256;
            int krow  = f4 >> 5;
            int colf4 = f4 & 31;
            float4 v = reinterpret_cast<const float4*>(W + (size_t)(kb + krow) * GC_OUT_DIM)[colf4];
            float* dst = &lw[krow * GC_WPAD + colf4 * 4];
            dst[0] = v.x; dst[1] = v.y; dst[2] = v.z; dst[3] = v.w;
        }
        __syncthreads();

        #pragma unroll
        for (int kk = 0; kk < 64; kk += 4) {
            const float* ax = &lx[l16 * GC_XPAD + kb + kk + 2 * half];
            v2f a; a.x = ax[0]; a.y = ax[1];
            v2f b;
            b.x = lw[(kk + 2 * half    ) * GC_WPAD + n0 + l16];
            b.y = lw[(kk + 2 * half + 1) * GC_WPAD + n0 + l16];
            acc = __builtin_amdgcn_wmma_f32_16x16x4_f32(
                false, a, false, b, (short)0, acc, false, false);
        }
    }

    #pragma unroll
    for (int i = 0; i < 8; ++i) {
        int m = i + 8 * half;
        if (r0 + m < nRows)
            xp[(size_t)(r0 + m) * GC_OUT_DIM + n0 + l16] = acc[i];
    }
}

// ---------------------------------------------------------------------------
// CSR construction
// ---------------------------------------------------------------------------
__global__ __launch_bounds__(256) void gc_zero_ints(int* __restrict__ p, int n) {
    int t = blockIdx.x * 256 + threadIdx.x;
    if (t < n) p[t] = 0;
}

__global__ __launch_bounds__(256) void gc_hist(const int* __restrict__ erow,
                                               int* __restrict__ cnt, int nEdges) {
    int e = blockIdx.x * 256 + threadIdx.x;
    if (e < nEdges) atomicAdd(&cnt[erow[e]], 1);
}

// Single-block exclusive scan of cnt[0..n) -> rowPtr[0..n], rowPtr[n] = total.
__global__ __launch_bounds__(1024) void gc_scan(const int* __restrict__ cnt,
                                                int* __restrict__ rowPtr, int n) {
    __shared__ int sums[1024];
    const int tid   = threadIdx.x;
    const int chunk = (n + 1023) / 1024;
    const int start = tid * chunk;
    const int end   = (start + chunk < n) ? (start + chunk) : n;

    int s = 0;
    for (int i = start; i < end; ++i) s += cnt[i];
    sums[tid] = s;
    __syncthreads();

    // Hillis-Steele inclusive scan over the 1024 chunk sums.
    for (int off = 1; off < 1024; off <<= 1) {
        int t = (tid >= off) ? sums[tid - off] : 0;
        __syncthreads();
        sums[tid] += t;
        __syncthreads();
    }

    int run = (tid == 0) ? 0 : sums[tid - 1];   // exclusive prefix of this chunk
    for (int i = start; i < end; ++i) { rowPtr[i] = run; run += cnt[i]; }
    if (tid == 1023) rowPtr[n] = sums[1023];
}

__global__ __launch_bounds__(256) void gc_scatter(const int* __restrict__ erow,
                                                  const int* __restrict__ ecol,
                                                  const float* __restrict__ eval,
                                                  const int* __restrict__ rowPtr,
                                                  int* __restrict__ fill,
                                                  int* __restrict__ csrCol,
                                                  float* __restrict__ csrVal,
                                                  int nEdges) {
    int e = blockIdx.x * 256 + threadIdx.x;
    if (e >= nEdges) return;
    int r   = erow[e];
    int pos = rowPtr[r] + atomicAdd(&fill[r], 1);
    csrCol[pos] = ecol[e];
    csrVal[pos] = eval[e];
}

// ---------------------------------------------------------------------------
// Row-parallel SpMM: one wave per row, lane L owns features [4L, 4L+4).
// Pure gathers from L2-resident xp; register accumulate; fused ReLU store.
// ---------------------------------------------------------------------------
__global__ __launch_bounds__(256) void gc_spmm_csr(const float* __restrict__ xp,
                                                   const int* __restrict__ rowPtr,
                                                   const int* __restrict__ csrCol,
                                                   const float* __restrict__ csrVal,
                                                   float* __restrict__ out,
                                                   int nRows) {
    int t    = blockIdx.x * 256 + threadIdx.x;
    int row  = t >> 5;
    int lane = t & 31;
    if (row >= nRows) return;

    const int beg = rowPtr[row];
    const int end = rowPtr[row + 1];

    float4 acc = make_float4(0.f, 0.f, 0.f, 0.f);
    for (int j = beg; j < end; ++j) {
        int   c = csrCol[j];          // wave-uniform -> broadcast from WGP$
        float v = csrVal[j];
        float4 m = reinterpret_cast<const float4*>(xp + (size_t)c * GC_OUT_DIM)[lane];
        acc.x = fmaf(m.x, v, acc.x);
        acc.y = fmaf(m.y, v, acc.y);
        acc.z = fmaf(m.z, v, acc.z);
        acc.w = fmaf(m.w, v, acc.w);
    }
    acc.x = fmaxf(acc.x, 0.f);
    acc.y = fmaxf(acc.y, 0.f);
    acc.z = fmaxf(acc.z, 0.f);
    acc.w = fmaxf(acc.w, 0.f);
    reinterpret_cast<float4*>(out + (size_t)row * GC_OUT_DIM)[lane] = acc;
}

// ---------------------------------------------------------------------------
// Fallback path (ws too small for CSR): edge-parallel f32 atomics.
// ---------------------------------------------------------------------------
__global__ __launch_bounds__(256) void gc_zero_f4(float4* __restrict__ out, int nF4) {
    int t = blockIdx.x * 256 + threadIdx.x;
    if (t < nF4) out[t] = make_float4(0.f, 0.f, 0.f, 0.f);
}

__global__ __launch_bounds__(256) void gc_spmm_atomic(const float* __restrict__ xp,
                                                      const int* __restrict__ erow,
                                                      const int* __restrict__ ecol,
                                                      const float* __restrict__ eval,
                                                      float* __restrict__ out,
                                                      int nEdges) {
    int t    = blockIdx.x * 256 + threadIdx.x;
    int e    = t >> 5;
    int lane = t & 31;
    if (e >= nEdges) return;
    int   r = erow[e];
    int   c = ecol[e];
    float v = eval[e];
    float4 m = reinterpret_cast<const float4*>(xp + (size_t)c * GC_OUT_DIM)[lane];
    float* dst = out + (size_t)r * GC_OUT_DIM + lane * 4;
    __hip_atomic_fetch_add(dst + 0, m.x * v, __ATOMIC_RELAXED, __HIP_MEMORY_SCOPE_AGENT);
    __hip_atomic_fetch_add(dst + 1, m.y * v, __ATOMIC_RELAXED, __HIP_MEMORY_SCOPE_AGENT);
    __hip_atomic_fetch_add(dst + 2, m.z * v, __ATOMIC_RELAXED, __HIP_MEMORY_SCOPE_AGENT);
    __hip_atomic_fetch_add(dst + 3, m.w * v, __ATOMIC_RELAXED, __HIP_MEMORY_SCOPE_AGENT);
}

__global__ __launch_bounds__(256) void gc_relu(float4* __restrict__ out, int nF4) {
    int t = blockIdx.x * 256 + threadIdx.x;
    if (t >= nF4) return;
    float4 v = out[t];
    v.x = fmaxf(v.x, 0.f); v.y = fmaxf(v.y, 0.f);
    v.z = fmaxf(v.z, 0.f); v.w = fmaxf(v.w, 0.f);
    out[t] = v;
}

static inline size_t gc_align256(size_t x) { return (x + 255) & ~(size_t)255; }

extern "C" void kernel_launch(void* const* d_in, const int* in_sizes, int n_in,
                              void* d_out, int out_size, void* d_ws, size_t ws_size,
                              hipStream_t stream) {
    const float* x    = (const float*)d_in[0];
    const float* W    = (const float*)d_in[1];
    const int*   erow = (const int*)d_in[2];
    const int*   ecol = (const int*)d_in[3];
    const float* eval = (const float*)d_in[4];
    float*       out  = (float*)d_out;

    const int nNodes = in_sizes[0] / GC_IN_DIM;   // 100000
    const int nEdges = in_sizes[2];               // 3200000
    const int nF4    = (nNodes * GC_OUT_DIM) / 4;

    // Workspace layout: xp | rowPtr | cnt+fill | csrCol | csrVal
    char* ws = (char*)d_ws;
    size_t off = 0;
    float* xp = (float*)(ws + off);
    off = gc_align256(off + (size_t)nNodes * GC_OUT_DIM * sizeof(float));
    size_t offRowPtr = off; off = gc_align256(off + (size_t)(nNodes + 1) * sizeof(int));
    size_t offCnt    = off; off = gc_align256(off + (size_t)2 * nNodes * sizeof(int));
    size_t offCol    = off; off = gc_align256(off + (size_t)nEdges * sizeof(int));
    size_t offVal    = off; off = gc_align256(off + (size_t)nEdges * sizeof(float));
    const bool useCSR = (ws_size >= off);

    // K1: dense projection (always).
    gc_gemm_wmma<<<(nNodes + 15) / 16, 256, 0, stream>>>(x, W, xp, nNodes);

    if (useCSR) {
        int*   rowPtr = (int*)(ws + offRowPtr);
        int*   cnt    = (int*)(ws + offCnt);          // [nNodes] histogram
        int*   fill   = cnt + nNodes;                 // [nNodes] scatter cursor
        int*   csrCol = (int*)(ws + offCol);
        float* csrVal = (float*)(ws + offVal);

        int n2 = 2 * nNodes;
        gc_zero_ints<<<(n2 + 255) / 256, 256, 0, stream>>>(cnt, n2);
        gc_hist<<<(nEdges + 255) / 256, 256, 0, stream>>>(erow, cnt, nEdges);
        gc_scan<<<1, 1024, 0, stream>>>(cnt, rowPtr, nNodes);
        gc_scatter<<<(nEdges + 255) / 256, 256, 0, stream>>>(
            erow, ecol, eval, rowPtr, fill, csrCol, csrVal, nEdges);

        long long thr = (long long)nNodes * 32;
        gc_spmm_csr<<<(int)((thr + 255) / 256), 256, 0, stream>>>(
            xp, rowPtr, csrCol, csrVal, out, nNodes);
    } else {
        gc_zero_f4<<<(nF4 + 255) / 256, 256, 0, stream>>>((float4*)out, nF4);
        long long thr = (long long)nEdges * 32;
        gc_spmm_atomic<<<(int)((thr + 255) / 256), 256, 0, stream>>>(
            xp, erow, ecol, eval, out, nEdges);
        gc_relu<<<(nF4 + 255) / 256, 256, 0, stream>>>((float4*)out, nF4);
    }
}